// MambaFusion_70076686402477
// MI455X (gfx1250) — compile-verified
//
#include <hip/hip_runtime.h>
#include <hip/hip_bf16.h>

typedef float v2f __attribute__((ext_vector_type(2)));
typedef float v8f __attribute__((ext_vector_type(8)));

#define WMMA_F32(a, b, c) \
    __builtin_amdgcn_wmma_f32_16x16x4_f32(false, (a), false, (b), (short)0, (c), false, false)

// Problem constants
#define BATCH    2
#define LT       512
#define LSEQ     2048            // 4*LT
#define DMODEL   512
#define DINNER   1024
#define ROWS     (BATCH * LSEQ)  // 4096
#define DTRANK   32
#define DSTATE   16

__device__ __forceinline__ float silu_f(float x) {
    return x / (1.0f + __expf(-x));
}
__device__ __forceinline__ float softplus_f(float x) {
    return (x > 20.0f) ? x : log1pf(__expf(x));
}

// ---------------------------------------------------------------------------
// GEMM 1: xz = gather(t0..t3) @ in_proj_w^T.   A: [4096 x 512] (gathered),
// W: [2048 x 512] row-major, out xz: [4096 x 2048].
// Gather: row r -> b=r/2048, l=r%2048, chunk c=l/128, j=l%128,
//         src tensor = c%4, src row = (c/4) + 4*j.
// 32x32 macro-tile per wave: 2 A frags x 2 B frags -> 4 independent WMMA chains.
// ---------------------------------------------------------------------------
__device__ __forceinline__ const float* gather_row_ptr(
        int r, const float* t0, const float* t1, const float* t2, const float* t3) {
    const int b = r >> 11, l = r & 2047;
    const int c = l >> 7, j = l & 127;
    const int tsel = c & 3, iph = c >> 2;
    const float* ts = (tsel == 0) ? t0 : (tsel == 1) ? t1 : (tsel == 2) ? t2 : t3;
    return ts + ((size_t)(b * LT + (iph + 4 * j))) * DMODEL;
}

__global__ void k_inproj(const float* __restrict__ t0, const float* __restrict__ t1,
                         const float* __restrict__ t2, const float* __restrict__ t3,
                         const float* __restrict__ W, float* __restrict__ xz) {
    constexpr int K = DMODEL, N = 2 * DINNER;
    const int lane = threadIdx.x & 31;
    const int wave = threadIdx.x >> 5;
    const int tile = blockIdx.x * (blockDim.x >> 5) + wave;
    const int nTN  = N / 32;
    const int m0   = (tile / nTN) * 32;
    const int n0   = (tile % nTN) * 32;
    const int half = lane >> 4, lr = lane & 15, kf = half * 2;

    const float* pA0 = gather_row_ptr(m0 + lr,      t0, t1, t2, t3) + kf;
    const float* pA1 = gather_row_ptr(m0 + 16 + lr, t0, t1, t2, t3) + kf;
    const float* pB0 = W + (size_t)(n0 + lr) * K + kf;
    const float* pB1 = W + (size_t)(n0 + 16 + lr) * K + kf;

    v8f a00 = {}, a01 = {}, a10 = {}, a11 = {};
#pragma unroll 4
    for (int k = 0; k < K; k += 4) {
        v2f fa0 = *(const v2f*)(pA0 + k);
        v2f fa1 = *(const v2f*)(pA1 + k);
        v2f fb0 = *(const v2f*)(pB0 + k);
        v2f fb1 = *(const v2f*)(pB1 + k);
        a00 = WMMA_F32(fa0, fb0, a00);
        a01 = WMMA_F32(fa0, fb1, a01);
        a10 = WMMA_F32(fa1, fb0, a10);
        a11 = WMMA_F32(fa1, fb1, a11);
    }
    float* pC0 = xz + (size_t)(m0 + half * 8) * N + n0 + lr;
    float* pC1 = xz + (size_t)(m0 + 16 + half * 8) * N + n0 + lr;
#pragma unroll
    for (int vr = 0; vr < 8; ++vr) {
        pC0[(size_t)vr * N]      = a00[vr];
        pC0[(size_t)vr * N + 16] = a01[vr];
        pC1[(size_t)vr * N]      = a10[vr];
        pC1[(size_t)vr * N + 16] = a11[vr];
    }
}

// ---------------------------------------------------------------------------
// Depthwise causal conv1d (K=4, left pad 3) + bias + SiLU over xc = xz[:, :1024]
// ---------------------------------------------------------------------------
__global__ void k_conv_silu(const float* __restrict__ xz, const float* __restrict__ cw,
                            const float* __restrict__ cb, float* __restrict__ xc) {
    const int idx = blockIdx.x * blockDim.x + threadIdx.x;  // over 4096*1024
    const int d   = idx & (DINNER - 1);
    const int row = idx >> 10;              // b*2048 + l
    const int l   = row & (LSEQ - 1);
    const float* base = xz + (size_t)row * (2 * DINNER) + d;
    const float* w    = cw + d * 4;
    float acc = cb[d];
#pragma unroll
    for (int jj = 0; jj < 4; ++jj) {
        const int ll = l + jj - 3;
        if (ll >= 0) acc += w[jj] * base[(long)(jj - 3) * (2 * DINNER)];
    }
    xc[idx] = silu_f(acc);
}

// ---------------------------------------------------------------------------
// GEMM 2: dbc = xc @ x_proj_w^T.  A: [4096 x 1024], W: [64 x 1024], out [4096 x 64]
// N is only 64, so keep a 16x32 macro-tile here.
// ---------------------------------------------------------------------------
__global__ void k_xproj(const float* __restrict__ A, const float* __restrict__ W,
                        float* __restrict__ dbc) {
    constexpr int K = DINNER, N = DTRANK + 2 * DSTATE;  // 64
    const int lane = threadIdx.x & 31;
    const int wave = threadIdx.x >> 5;
    const int tile = blockIdx.x * (blockDim.x >> 5) + wave;
    const int nTN  = N / 32;
    const int m0   = (tile / nTN) * 16;
    const int n0   = (tile % nTN) * 32;
    const int half = lane >> 4, lr = lane & 15, kf = half * 2;

    const float* pA  = A + (size_t)(m0 + lr) * K + kf;
    const float* pB0 = W + (size_t)(n0 + lr) * K + kf;
    const float* pB1 = W + (size_t)(n0 + 16 + lr) * K + kf;
    v8f acc0 = {}, acc1 = {};
#pragma unroll 4
    for (int k = 0; k < K; k += 4) {
        v2f a  = *(const v2f*)(pA + k);
        v2f b0 = *(const v2f*)(pB0 + k);
        v2f b1 = *(const v2f*)(pB1 + k);
        acc0 = WMMA_F32(a, b0, acc0);
        acc1 = WMMA_F32(a, b1, acc1);
    }
    float* pC = dbc + (size_t)(m0 + half * 8) * N + n0 + lr;
#pragma unroll
    for (int vr = 0; vr < 8; ++vr) {
        pC[(size_t)vr * N]      = acc0[vr];
        pC[(size_t)vr * N + 16] = acc1[vr];
    }
}

// ---------------------------------------------------------------------------
// GEMM 3: delta = softplus(dbc[:, :32] @ dt_w^T + dt_b).
// A: [4096 x 32] (lda=64), W: [1024 x 32], out [4096 x 1024].  32x32 tiles.
// ---------------------------------------------------------------------------
__global__ void k_dtproj(const float* __restrict__ dbc, const float* __restrict__ W,
                         const float* __restrict__ dtb, float* __restrict__ delta) {
    constexpr int K = DTRANK, N = DINNER, LDA = DTRANK + 2 * DSTATE;
    const int lane = threadIdx.x & 31;
    const int wave = threadIdx.x >> 5;
    const int tile = blockIdx.x * (blockDim.x >> 5) + wave;
    const int nTN  = N / 32;
    const int m0   = (tile / nTN) * 32;
    const int n0   = (tile % nTN) * 32;
    const int half = lane >> 4, lr = lane & 15, kf = half * 2;

    const float* pA0 = dbc + (size_t)(m0 + lr) * LDA + kf;
    const float* pA1 = dbc + (size_t)(m0 + 16 + lr) * LDA + kf;
    const float* pB0 = W + (size_t)(n0 + lr) * K + kf;
    const float* pB1 = W + (size_t)(n0 + 16 + lr) * K + kf;
    v8f a00 = {}, a01 = {}, a10 = {}, a11 = {};
#pragma unroll
    for (int k = 0; k < K; k += 4) {
        v2f fa0 = *(const v2f*)(pA0 + k);
        v2f fa1 = *(const v2f*)(pA1 + k);
        v2f fb0 = *(const v2f*)(pB0 + k);
        v2f fb1 = *(const v2f*)(pB1 + k);
        a00 = WMMA_F32(fa0, fb0, a00);
        a01 = WMMA_F32(fa0, fb1, a01);
        a10 = WMMA_F32(fa1, fb0, a10);
        a11 = WMMA_F32(fa1, fb1, a11);
    }
    const float bias0 = dtb[n0 + lr];
    const float bias1 = dtb[n0 + 16 + lr];
    float* pC0 = delta + (size_t)(m0 + half * 8) * N + n0 + lr;
    float* pC1 = delta + (size_t)(m0 + 16 + half * 8) * N + n0 + lr;
#pragma unroll
    for (int vr = 0; vr < 8; ++vr) {
        pC0[(size_t)vr * N]      = softplus_f(a00[vr] + bias0);
        pC0[(size_t)vr * N + 16] = softplus_f(a01[vr] + bias1);
        pC1[(size_t)vr * N]      = softplus_f(a10[vr] + bias0);
        pC1[(size_t)vr * N + 16] = softplus_f(a11[vr] + bias1);
    }
}

// ---------------------------------------------------------------------------
// Selective scan: h_t = exp(delta*A) h_{t-1} + (delta*xc) B_t ; y = <h,C> + D*xc
// One lane per (b, d) channel, 16 states in registers. B_t/C_t are d-invariant
// so they are staged once per chunk in LDS and shared by the whole block.
// ---------------------------------------------------------------------------
#define SCAN_THREADS 64
#define SCAN_CHUNK   128
__global__ void k_scan(const float* __restrict__ delta, const float* __restrict__ xc,
                       const float* __restrict__ dbc, const float* __restrict__ A_log,
                       const float* __restrict__ D_param, float* __restrict__ y) {
    __shared__ float sB[SCAN_CHUNK][DSTATE];
    __shared__ float sC[SCAN_CHUNK][DSTATE];
    const int b  = blockIdx.x / (DINNER / SCAN_THREADS);
    const int d  = (blockIdx.x % (DINNER / SCAN_THREADS)) * SCAN_THREADS + threadIdx.x;

    float A[DSTATE];
#pragma unroll
    for (int s = 0; s < DSTATE; ++s) A[s] = -__expf(A_log[d * DSTATE + s]);
    const float Dv = D_param[d];
    float h[DSTATE];
#pragma unroll
    for (int s = 0; s < DSTATE; ++s) h[s] = 0.0f;

    for (int l0 = 0; l0 < LSEQ; l0 += SCAN_CHUNK) {
        __syncthreads();
        for (int t = threadIdx.x; t < SCAN_CHUNK * DSTATE; t += SCAN_THREADS) {
            const int li = t >> 4, s = t & 15;
            const float* rowp = dbc + (size_t)((b << 11) + l0 + li) * 64;
            sB[li][s] = rowp[DTRANK + s];
            sC[li][s] = rowp[DTRANK + DSTATE + s];
        }
        __syncthreads();
        for (int li = 0; li < SCAN_CHUNK; ++li) {
            const size_t ro = (size_t)((b << 11) + l0 + li) * DINNER + d;
            const float dl = delta[ro];
            const float xv = xc[ro];
            const float dx = dl * xv;
            float accv = Dv * xv;
#pragma unroll
            for (int s = 0; s < DSTATE; ++s) {
                h[s] = __expf(dl * A[s]) * h[s] + dx * sB[li][s];
                accv += h[s] * sC[li][s];
            }
            y[ro] = accv;
        }
    }
}

// ---------------------------------------------------------------------------
// Gating: g = y * silu(z),  z = xz[:, 1024:]
// ---------------------------------------------------------------------------
__global__ void k_gate(const float* __restrict__ y, const float* __restrict__ xz,
                       float* __restrict__ g) {
    const int idx = blockIdx.x * blockDim.x + threadIdx.x;  // over 4096*1024
    const int d   = idx & (DINNER - 1);
    const int row = idx >> 10;
    const float z = xz[(size_t)row * (2 * DINNER) + DINNER + d];
    g[idx] = y[idx] * silu_f(z);
}

// ---------------------------------------------------------------------------
// GEMM 4: out = g @ out_proj_w^T, fused de-interleave scatter + residual add.
// A: [4096 x 1024], W: [512 x 1024]; row m -> tensor i=(m%2048)&3, step k=(m%2048)>>2
// 32x32 macro-tiles.
// ---------------------------------------------------------------------------
__global__ void k_outproj(const float* __restrict__ A, const float* __restrict__ W,
                          const float* __restrict__ t0, const float* __restrict__ t1,
                          const float* __restrict__ t2, const float* __restrict__ t3,
                          float* __restrict__ out) {
    constexpr int K = DINNER, N = DMODEL;
    const int lane = threadIdx.x & 31;
    const int wave = threadIdx.x >> 5;
    const int tile = blockIdx.x * (blockDim.x >> 5) + wave;
    const int nTN  = N / 32;
    const int m0   = (tile / nTN) * 32;
    const int n0   = (tile % nTN) * 32;
    const int half = lane >> 4, lr = lane & 15, kf = half * 2;

    const float* pA0 = A + (size_t)(m0 + lr) * K + kf;
    const float* pA1 = A + (size_t)(m0 + 16 + lr) * K + kf;
    const float* pB0 = W + (size_t)(n0 + lr) * K + kf;
    const float* pB1 = W + (size_t)(n0 + 16 + lr) * K + kf;
    v8f a00 = {}, a01 = {}, a10 = {}, a11 = {};
#pragma unroll 4
    for (int k = 0; k < K; k += 4) {
        v2f fa0 = *(const v2f*)(pA0 + k);
        v2f fa1 = *(const v2f*)(pA1 + k);
        v2f fb0 = *(const v2f*)(pB0 + k);
        v2f fb1 = *(const v2f*)(pB1 + k);
        a00 = WMMA_F32(fa0, fb0, a00);
        a01 = WMMA_F32(fa0, fb1, a01);
        a10 = WMMA_F32(fa1, fb0, a10);
        a11 = WMMA_F32(fa1, fb1, a11);
    }
#pragma unroll
    for (int sub = 0; sub < 2; ++sub) {
#pragma unroll
        for (int vr = 0; vr < 8; ++vr) {
            const int m = m0 + sub * 16 + vr + half * 8;
            const int b = m >> 11, l = m & 2047;
            const int i = l & 3, kk = l >> 2;
            const float* ts = (i == 0) ? t0 : (i == 1) ? t1 : (i == 2) ? t2 : t3;
            const size_t off = (size_t)b * (LT * DMODEL) + (size_t)kk * DMODEL + n0 + lr;
            const float c0 = (sub == 0) ? a00[vr] : a10[vr];
            const float c1 = (sub == 0) ? a01[vr] : a11[vr];
            out[(size_t)i * (BATCH * LT * DMODEL) + off]      = c0 + ts[off];
            out[(size_t)i * (BATCH * LT * DMODEL) + off + 16] = c1 + ts[off + 16];
        }
    }
}

// ---------------------------------------------------------------------------
extern "C" void kernel_launch(void* const* d_in, const int* in_sizes, int n_in,
                              void* d_out, int out_size, void* d_ws, size_t ws_size,
                              hipStream_t stream) {
    const float* t0   = (const float*)d_in[0];
    const float* t1   = (const float*)d_in[1];
    const float* t2   = (const float*)d_in[2];
    const float* t3   = (const float*)d_in[3];
    const float* inW  = (const float*)d_in[4];
    const float* cw   = (const float*)d_in[5];
    const float* cb   = (const float*)d_in[6];
    const float* xpW  = (const float*)d_in[7];
    const float* dtW  = (const float*)d_in[8];
    const float* dtB  = (const float*)d_in[9];
    const float* Alog = (const float*)d_in[10];
    const float* Dp   = (const float*)d_in[11];
    const float* outW = (const float*)d_in[12];
    float* out = (float*)d_out;

    // Workspace layout (floats)
    float* ws    = (float*)d_ws;
    float* xz    = ws;                                   // [4096 x 2048]
    float* xc    = xz    + (size_t)ROWS * 2 * DINNER;    // [4096 x 1024]
    float* dbc   = xc    + (size_t)ROWS * DINNER;        // [4096 x 64]
    float* delta = dbc   + (size_t)ROWS * 64;            // [4096 x 1024]
    float* yb    = delta + (size_t)ROWS * DINNER;        // [4096 x 1024]
    float* g     = delta;                                // reuse (delta dead after scan)

    const int THR = 256, WPB = THR / 32;

    // 1) in_proj GEMM with fused interleave-gather: (4096/32) x (2048/32) = 128x64 tiles
    k_inproj<<<(128 * 64) / WPB, THR, 0, stream>>>(t0, t1, t2, t3, inW, xz);
    // 2) depthwise causal conv + SiLU
    k_conv_silu<<<(ROWS * DINNER) / THR, THR, 0, stream>>>(xz, cw, cb, xc);
    // 3) x_proj GEMM: (4096/16) x (64/32) = 256x2 tiles
    k_xproj<<<(256 * 2) / WPB, THR, 0, stream>>>(xc, xpW, dbc);
    // 4) dt GEMM + softplus: (4096/32) x (1024/32) = 128x32 tiles
    k_dtproj<<<(128 * 32) / WPB, THR, 0, stream>>>(dbc, dtW, dtB, delta);
    // 5) selective scan (sequential in L, parallel over b x d)
    k_scan<<<BATCH * (DINNER / SCAN_THREADS), SCAN_THREADS, 0, stream>>>(
        delta, xc, dbc, Alog, Dp, yb);
    // 6) gating
    k_gate<<<(ROWS * DINNER) / THR, THR, 0, stream>>>(yb, xz, g);
    // 7) out_proj GEMM + residual scatter: (4096/32) x (512/32) = 128x16 tiles
    k_outproj<<<(128 * 16) / WPB, THR, 0, stream>>>(g, outW, t0, t1, t2, t3, out);
}